// AlignModel_64407329571659
// MI455X (gfx1250) — compile-verified
//
#include <hip/hip_runtime.h>

typedef __attribute__((ext_vector_type(2))) float v2f;
typedef __attribute__((ext_vector_type(4))) float v4f;
typedef __attribute__((ext_vector_type(8))) float v8f;

#define NB    8
#define SENTS 2048
#define DDIM  512
// flattened rows of prop_state: NB * 2*SENTS = 32768, in 2048 tiles of 16 rows
#define NTILES 2048
#define WAVES_PER_BLK 8

// ---------------------------------------------------------------------------
// Phase 1: per-row dot products via V_WMMA_F32_16X16X4_F32.
// One wave32 per 16-row tile. A-frag (16x4 f32) per ISA layout:
//   lanes 0-15 : VGPR0=K0, VGPR1=K1 ; lanes 16-31 : VGPR0=K2, VGPR1=K3
// B-frag (4x16) broadcasts w[k] across all 16 columns with the mirrored
// K layout, so every column of D holds the same 16 dot products.
// ---------------------------------------------------------------------------
__global__ __launch_bounds__(32 * WAVES_PER_BLK) void
gemv_scores_wmma(const float* __restrict__ X, const float* __restrict__ W,
                 float* __restrict__ S) {
  const int lane = threadIdx.x & 31;
  const int wave = threadIdx.x >> 5;
  const int tile = blockIdx.x * WAVES_PER_BLK + wave;   // 0..NTILES-1
  const int row0 = tile << 4;                           // first of 16 rows
  const int half = (row0 >> 11) & 1;                    // 0 = orig/w_o, 1 = simp/w_s

  const int m    = lane & 15;                           // row within tile
  const int koff = (lane >> 4) << 1;                    // 0 or 2

  const float* arow = X + (size_t)(row0 + m) * DDIM + koff;
  const float* wrow = W + half * DDIM + koff;

  v8f acc0 = {};
  v8f acc1 = {};
#pragma unroll 4
  for (int k = 0; k < DDIM; k += 8) {
    v2f a0 = *(const v2f*)(arow + k);
    v2f b0 = *(const v2f*)(wrow + k);
    v2f a1 = *(const v2f*)(arow + k + 4);
    v2f b1 = *(const v2f*)(wrow + k + 4);
    acc0 = __builtin_amdgcn_wmma_f32_16x16x4_f32(false, a0, false, b0,
                                                 (short)0, acc0, false, false);
    acc1 = __builtin_amdgcn_wmma_f32_16x16x4_f32(false, a1, false, b1,
                                                 (short)0, acc1, false, false);
  }

  // C/D layout: lane 0 VGPR r = D[M=r,N=0]; lane 16 VGPR r = D[M=8+r,N=0].
  // All columns identical -> lanes 0 and 16 write the 16 row scores.
  if ((lane & 15) == 0) {
    float* out = S + row0 + ((lane >> 4) << 3);
#pragma unroll
    for (int r = 0; r < 8; ++r) out[r] = acc0[r] + acc1[r];
  }
}

// ---------------------------------------------------------------------------
// Phase 2: out[b,i,j] = sigmoid(s_s[b,i] + s_o[b,j] + bias)
// One 256-thread block per (b,i) row; 2048 floats written per block with
// non-temporal float4 stores (output is write-once, never re-read).
// ---------------------------------------------------------------------------
__global__ __launch_bounds__(256) void
outer_sigmoid(const float* __restrict__ S, const float* __restrict__ bias,
              float* __restrict__ out) {
  const int i = blockIdx.x;   // 0..SENTS-1
  const int b = blockIdx.y;   // 0..NB-1

  const float base = S[b * 2 * SENTS + SENTS + i] + bias[0];
  const float* so  = S + b * 2 * SENTS;          // s_o row for this batch
  float* orow = out + ((size_t)(b * SENTS + i) << 11);

#pragma unroll
  for (int c = 0; c < 2; ++c) {
    const int j = (threadIdx.x << 2) + (c << 10);
    v4f s4 = *(const v4f*)(so + j);
    v4f o;
    o.x = __builtin_amdgcn_rcpf(1.0f + __expf(-(base + s4.x)));
    o.y = __builtin_amdgcn_rcpf(1.0f + __expf(-(base + s4.y)));
    o.z = __builtin_amdgcn_rcpf(1.0f + __expf(-(base + s4.z)));
    o.w = __builtin_amdgcn_rcpf(1.0f + __expf(-(base + s4.w)));
    __builtin_nontemporal_store(o, (v4f*)(orow + j));
  }
}

extern "C" void kernel_launch(void* const* d_in, const int* in_sizes, int n_in,
                              void* d_out, int out_size, void* d_ws, size_t ws_size,
                              hipStream_t stream) {
  // setup_inputs order: A (unused zeros), prop_state, W, b
  const float* X    = (const float*)d_in[1];   // (8, 4096, 512) f32
  const float* W    = (const float*)d_in[2];   // (1, 1024) f32
  const float* bias = (const float*)d_in[3];   // (1,) f32
  float* out = (float*)d_out;                  // (8, 2048, 2048) f32
  float* S   = (float*)d_ws;                   // 32768 f32 scores (128 KB)

  gemv_scores_wmma<<<dim3(NTILES / WAVES_PER_BLK), dim3(32 * WAVES_PER_BLK), 0,
                     stream>>>(X, W, S);
  outer_sigmoid<<<dim3(SENTS, NB), dim3(256), 0, stream>>>(S, bias, out);
}